// TokenLabelOfflineEpisodicReweight_61280593379833
// MI455X (gfx1250) — compile-verified
//
#include <hip/hip_runtime.h>
#include <math.h>

#define B_ 2
#define Q_ 75
#define N_ 5
#define T_ 196
#define C_ 384
#define H_ 1536
#define TEMP_ 10.0f
#define PAD_ 6          // padded minor stride for mapq/wcwtp; column 5 is zero

typedef __attribute__((ext_vector_type(2))) float v2f;
typedef __attribute__((ext_vector_type(8))) float v8f;
typedef int v4i_vs __attribute__((vector_size(16)));   // matches builtin param type

#if defined(__HIP_DEVICE_COMPILE__) && defined(__gfx1250__) && \
    __has_builtin(__builtin_amdgcn_wmma_f32_16x16x4_f32)
#define USE_WMMA_F32 1
#else
#define USE_WMMA_F32 0
#endif

#if defined(__HIP_DEVICE_COMPILE__) && defined(__gfx1250__) && \
    __has_builtin(__builtin_amdgcn_global_load_async_to_lds_b128) && \
    __has_builtin(__builtin_amdgcn_s_wait_asynccnt)
#define USE_ASYNC_LDS 1
#else
#define USE_ASYNC_LDS 0
#endif

#define AS1 __attribute__((address_space(1)))
#define AS3 __attribute__((address_space(3)))

__device__ __forceinline__ float wave_reduce_sum(float v) {
    #pragma unroll
    for (int off = 16; off > 0; off >>= 1) v += __shfl_xor(v, off, 32);
    return v;
}

__device__ __forceinline__ float sigmoidf_(float x) {
    return 1.0f / (1.0f + expf(-x));
}

// ---------------------------------------------------------------------------
// K1: mean over T for each (b,n,c)   feat_shot (B,N,1,T,C) -> mean_cls (B,N,C)
// ---------------------------------------------------------------------------
__global__ void k_mean_cls(const float* __restrict__ feat_shot,
                           float* __restrict__ mean_cls) {
    int bn = blockIdx.x;           // 0..B*N-1
    int c  = threadIdx.x;          // 0..C-1
    const float* p = feat_shot + (size_t)bn * T_ * C_ + c;
    float acc = 0.0f;
    for (int t = 0; t < T_; ++t) acc += p[(size_t)t * C_];
    mean_cls[bn * C_ + c] = acc * (1.0f / T_);
}

// ---------------------------------------------------------------------------
// K2: mean over N  -> mean_task (B,C)
// ---------------------------------------------------------------------------
__global__ void k_mean_task(const float* __restrict__ mean_cls,
                            float* __restrict__ mean_task) {
    int b = blockIdx.x;
    int c = threadIdx.x;
    float acc = 0.0f;
    for (int n = 0; n < N_; ++n) acc += mean_cls[(b * N_ + n) * C_ + c];
    mean_task[b * C_ + c] = acc * (1.0f / N_);
}

// ---------------------------------------------------------------------------
// K3: one MLP row per block: out = (relu(x@w1+b1))@w2+b2, optional sigmoid
// ---------------------------------------------------------------------------
__global__ void k_mlp(const float* __restrict__ in,
                      const float* __restrict__ w1, const float* __restrict__ b1,
                      const float* __restrict__ w2, const float* __restrict__ b2,
                      float* __restrict__ out, int do_sigmoid) {
    __shared__ float inrow[C_];
    __shared__ float hidden[H_];
    int row = blockIdx.x;
    int tid = threadIdx.x;         // 512
    if (tid < C_) inrow[tid] = in[row * C_ + tid];
    __syncthreads();
    for (int j = tid; j < H_; j += 512) {
        float acc = b1[j];
        for (int c = 0; c < C_; ++c) acc = fmaf(inrow[c], w1[c * H_ + j], acc);
        hidden[j] = fmaxf(acc, 0.0f);
    }
    __syncthreads();
    if (tid < C_) {
        float acc = b2[tid];
        for (int j = 0; j < H_; ++j) acc = fmaf(hidden[j], w2[j * C_ + tid], acc);
        if (do_sigmoid) acc = sigmoidf_(acc);
        out[row * C_ + tid] = acc;
    }
}

// ---------------------------------------------------------------------------
// K4: shot-side per (b,n): ns[t]=||fs_t||, map_s, xs_rw (l2norm), xsn (protos)
// ---------------------------------------------------------------------------
__global__ void k_shot(const float* __restrict__ feat_shot,
                       const float* __restrict__ x_shot,
                       const float* __restrict__ wt,
                       const float* __restrict__ wc,
                       float* __restrict__ ns,
                       float* __restrict__ xs_rw,
                       float* __restrict__ xsn) {
    int bn = blockIdx.x;           // 0..B*N-1
    int b  = bn / N_;
    int tid = threadIdx.x, lane = tid & 31, wave = tid >> 5;
    __shared__ float wcwt[C_];
    __shared__ float maps[T_];
    __shared__ float red[2];
    const float* fsrow = feat_shot + (size_t)bn * T_ * C_;
    if (tid < C_) wcwt[tid] = wc[bn * C_ + tid] * wt[b * C_ + tid];
    if (tid < 2) red[tid] = 0.0f;
    __syncthreads();
    for (int t = wave; t < T_; t += 16) {
        const float* r = fsrow + (size_t)t * C_;
        float ss = 0.0f, g = 0.0f;
        for (int c = lane; c < C_; c += 32) {
            float v = r[c];
            ss = fmaf(v, v, ss);
            g  = fmaf(v, wcwt[c], g);
        }
        ss = wave_reduce_sum(ss);
        g  = wave_reduce_sum(g);
        if (lane == 0) {
            ns[bn * T_ + t] = sqrtf(ss);
            maps[t] = sigmoidf_(g);
        }
    }
    __syncthreads();
    float val = 0.0f, xv = 0.0f;
    if (tid < C_) {
        float acc = 0.0f;
        for (int t = 0; t < T_; ++t)
            acc = fmaf(fsrow[(size_t)t * C_ + tid], maps[t], acc);
        val = wc[bn * C_ + tid] * acc * (1.0f / T_);
        xv  = x_shot[bn * C_ + tid];       // K==1: mean over K is identity
        atomicAdd(&red[0], val * val);
        atomicAdd(&red[1], xv * xv);
    }
    __syncthreads();
    if (tid < C_) {
        xs_rw[bn * C_ + tid] = val / (1e-16f + sqrtf(red[0]));
        xsn[bn * C_ + tid]   = xv / fmaxf(sqrtf(red[1]), 1e-12f);
    }
}

// ---------------------------------------------------------------------------
// K5: main fused per-(b,q) kernel. 512 threads, dyn LDS:
//   fq[T*C] | mapq[T*PAD] | wcwtp[C*PAD] | wmax[16*N] | ssdp[2*N]
// ---------------------------------------------------------------------------
__global__ void k_main(const float* __restrict__ feat_shot,
                       const float* __restrict__ feat_query,
                       const float* __restrict__ x_query,
                       const float* __restrict__ wt,
                       const float* __restrict__ wc,
                       const float* __restrict__ ns,
                       const float* __restrict__ xs_rw,
                       const float* __restrict__ xsn,
                       float* __restrict__ out) {
    extern __shared__ float sm[];
    float* fq    = sm;                    // T_*C_
    float* mapq  = fq    + T_ * C_;       // T_*PAD_   ([t][n], col 5 == 0)
    float* wcwtp = mapq  + T_ * PAD_;     // C_*PAD_   ([c][n], col 5 == 0)
    float* wmax  = wcwtp + C_ * PAD_;     // 16*N_
    float* ssdp  = wmax  + 16 * N_;       // 2*N_  (ss then dp)

    int bq = blockIdx.x;                  // 0..B*Q-1
    int b  = bq / Q_;
    int tid = threadIdx.x, lane = tid & 31, wave = tid >> 5;

    // stage 0: fill LDS (async global->LDS path on CDNA5, bypasses VGPRs)
    const float* fqg = feat_query + (size_t)bq * T_ * C_;
#if USE_ASYNC_LDS
    {
        AS1 v4i_vs* g = (AS1 v4i_vs*)(uintptr_t)fqg;
        AS3 v4i_vs* l = (AS3 v4i_vs*)(unsigned int)(uintptr_t)fq;
        for (int i = tid; i < (T_ * C_ * 4) / 16; i += 512)
            __builtin_amdgcn_global_load_async_to_lds_b128(g + i, l + i, 0, 0);
        __builtin_amdgcn_s_wait_asynccnt(0);
    }
#else
    {
        const float4* src4 = (const float4*)fqg;
        float4*       dst4 = (float4*)fq;
        for (int i = tid; i < (T_ * C_) / 4; i += 512) dst4[i] = src4[i];
    }
#endif
    if (tid < C_) {
        float w = wt[b * C_ + tid];
        #pragma unroll
        for (int n = 0; n < N_; ++n)
            wcwtp[tid * PAD_ + n] = wc[(b * N_ + n) * C_ + tid] * w;
        wcwtp[tid * PAD_ + N_] = 0.0f;    // zero column
    }
    if (tid < T_) mapq[tid * PAD_ + N_] = 0.0f;  // zero column
    if (tid < 2 * N_) ssdp[tid] = 0.0f;
    __syncthreads();

    // stage 1: per-t cosine sims (VALU wave reductions), running max over t
    float lmax[N_];
    #pragma unroll
    for (int n = 0; n < N_; ++n) lmax[n] = -3.0e38f;
    for (int t = wave; t < T_; t += 16) {
        const float* fr  = feat_shot + ((size_t)b * N_ * T_ + t) * C_;
        const float* fqr = fq + (size_t)t * C_;
        float ss = 0.0f;
        float dn[N_];
#if !USE_WMMA_F32
        float gn[N_];
#endif
        #pragma unroll
        for (int n = 0; n < N_; ++n) {
            dn[n] = 0.0f;
#if !USE_WMMA_F32
            gn[n] = 0.0f;
#endif
        }
        for (int c = lane; c < C_; c += 32) {
            float v = fqr[c];
            ss = fmaf(v, v, ss);
            #pragma unroll
            for (int n = 0; n < N_; ++n) {
                dn[n] = fmaf(v, fr[(size_t)n * T_ * C_ + c], dn[n]);
#if !USE_WMMA_F32
                gn[n] = fmaf(v, wcwtp[c * PAD_ + n], gn[n]);
#endif
            }
        }
        ss = wave_reduce_sum(ss);
        #pragma unroll
        for (int n = 0; n < N_; ++n) {
            dn[n] = wave_reduce_sum(dn[n]);
#if !USE_WMMA_F32
            gn[n] = wave_reduce_sum(gn[n]);
#endif
        }
        if (lane == 0) {
            float nqv = sqrtf(ss);
            #pragma unroll
            for (int n = 0; n < N_; ++n) {
                float denom = fmaxf(nqv * ns[(b * N_ + n) * T_ + t], 1e-8f);
                lmax[n] = fmaxf(lmax[n], dn[n] / denom);
#if !USE_WMMA_F32
                mapq[t * PAD_ + n] = sigmoidf_(gn[n]);
#endif
            }
        }
    }
    if (lane == 0) {
        #pragma unroll
        for (int n = 0; n < N_; ++n) wmax[wave * N_ + n] = lmax[n];
    }

#if USE_WMMA_F32
    // stage 2 (WMMA): g[t,n] = fq(196x384) @ wcwt(384x16pad) -> map_q
    // Unconditional fragment loads: lanes n>=5 read the hardwired zero column.
    for (int tile = wave; tile < 13; tile += 16) {
        int t0 = tile * 16;
        int m  = lane & 15, hi = lane >> 4;
        int ms = (m < N_) ? m : N_;                        // zero column for n>=5
        int row = t0 + m; if (row > T_ - 1) row = T_ - 1;  // tail clamp
        const float* arow = fq + (size_t)row * C_;
        v8f acc = {0.f, 0.f, 0.f, 0.f, 0.f, 0.f, 0.f, 0.f};
        for (int k0 = 0; k0 < C_; k0 += 4) {
            int ka = k0 + 2 * hi;
            v2f a, bb;
            a.x  = arow[ka];
            a.y  = arow[ka + 1];
            bb.x = wcwtp[ka * PAD_ + ms];
            bb.y = wcwtp[(ka + 1) * PAD_ + ms];
            acc = __builtin_amdgcn_wmma_f32_16x16x4_f32(
                false, a, false, bb, (short)0, acc, false, false);
        }
        union { v8f v; float f[8]; } u; u.v = acc;
        if (m < N_) {
            #pragma unroll
            for (int j = 0; j < 8; ++j) {
                int t = t0 + j + 8 * hi;
                if (t < T_) mapq[t * PAD_ + m] = sigmoidf_(u.f[j]);
            }
        }
    }
#endif
    __syncthreads();

    // stage 3: xq_acc[n,c] = sum_t mapq[t,n]*fq[t,c]; fold into ss/dp sums
#if USE_WMMA_F32
    for (int tile = wave; tile < 24; tile += 16) {
        int c0 = tile * 16;
        int m  = lane & 15, hi = lane >> 4;
        int ms = (m < N_) ? m : N_;                        // zero column for n>=5
        v8f acc = {0.f, 0.f, 0.f, 0.f, 0.f, 0.f, 0.f, 0.f};
        for (int t0 = 0; t0 < T_; t0 += 4) {      // 49 K-steps, covers t=0..195
            int ta = t0 + 2 * hi;
            v2f a, bb;
            a.x  = mapq[ta * PAD_ + ms];
            a.y  = mapq[(ta + 1) * PAD_ + ms];
            bb.x = fq[(size_t)ta * C_ + c0 + m];
            bb.y = fq[(size_t)(ta + 1) * C_ + c0 + m];
            acc = __builtin_amdgcn_wmma_f32_16x16x4_f32(
                false, a, false, bb, (short)0, acc, false, false);
        }
        union { v8f v; float f[8]; } u; u.v = acc;
        int c = c0 + m;
        #pragma unroll
        for (int j = 0; j < N_; ++j) {            // n=j lives in lanes 0..15
            // unconditional loads (same address for both lane halves), then select
            float wcv = wc[(b * N_ + j) * C_ + c];
            float xrv = xs_rw[(b * N_ + j) * C_ + c];
            float val = (hi == 0) ? wcv * u.f[j] * (1.0f / T_) : 0.0f;
            float ssv = val * val;
            float dpv = val * xrv;
            ssv = wave_reduce_sum(ssv);
            dpv = wave_reduce_sum(dpv);
            if (lane == 0) {
                atomicAdd(&ssdp[j], ssv);
                atomicAdd(&ssdp[N_ + j], dpv);
            }
        }
    }
#else
    for (int idx = tid; idx < N_ * C_; idx += 512) {
        int n = idx / C_, c = idx % C_;
        float acc = 0.0f;
        for (int t = 0; t < T_; ++t)
            acc = fmaf(fq[(size_t)t * C_ + c], mapq[t * PAD_ + n], acc);
        float val = wc[(b * N_ + n) * C_ + c] * acc * (1.0f / T_);
        atomicAdd(&ssdp[n], val * val);
        atomicAdd(&ssdp[N_ + n], val * xs_rw[(b * N_ + n) * C_ + c]);
    }
#endif

    // stage 4: cls_logits (wave 0): 10 * normalize(x_query) . xsn
    if (wave == 0) {
        const float* xqr = x_query + (size_t)bq * C_;
        float ssq = 0.0f;
        float dnn[N_];
        #pragma unroll
        for (int n = 0; n < N_; ++n) dnn[n] = 0.0f;
        for (int c = lane; c < C_; c += 32) {
            float v = xqr[c];
            ssq = fmaf(v, v, ssq);
            #pragma unroll
            for (int n = 0; n < N_; ++n)
                dnn[n] = fmaf(v, xsn[(b * N_ + n) * C_ + c], dnn[n]);
        }
        ssq = wave_reduce_sum(ssq);
        #pragma unroll
        for (int n = 0; n < N_; ++n) dnn[n] = wave_reduce_sum(dnn[n]);
        if (lane == 0) {
            float inv = 1.0f / fmaxf(sqrtf(ssq), 1e-12f);
            #pragma unroll
            for (int n = 0; n < N_; ++n)
                out[300 + bq * N_ + n] = TEMP_ * dnn[n] * inv;
        }
    }
    __syncthreads();

    // stage 5: finals
    if (tid < N_) {
        float cr = ssdp[N_ + tid] / (1e-16f + sqrtf(ssdp[tid]));
        out[1050 + bq * N_ + tid] = cr;
    }
    if (tid == 0) {
        float s = 0.0f;
        for (int n = 0; n < N_; ++n) {
            float m = -3.0e38f;
            for (int w = 0; w < 16; ++w) m = fmaxf(m, wmax[w * N_ + n]);
            s += m;
        }
        float lg = s * (1.0f / N_);
        out[bq] = lg;             // logits
        out[150 + bq] = lg;       // logits_reweight (identical by reference)
    }
}

// ---------------------------------------------------------------------------
extern "C" void kernel_launch(void* const* d_in, const int* in_sizes, int n_in,
                              void* d_out, int out_size, void* d_ws, size_t ws_size,
                              hipStream_t stream) {
    const float* feat_shot  = (const float*)d_in[0];
    const float* feat_query = (const float*)d_in[1];
    const float* x_shot     = (const float*)d_in[2];
    const float* x_query    = (const float*)d_in[3];
    const float* w1_task    = (const float*)d_in[4];
    const float* b1_task    = (const float*)d_in[5];
    const float* w2_task    = (const float*)d_in[6];
    const float* b2_task    = (const float*)d_in[7];
    const float* w1_cls     = (const float*)d_in[8];
    const float* b1_cls     = (const float*)d_in[9];
    const float* w2_cls     = (const float*)d_in[10];
    const float* b2_cls     = (const float*)d_in[11];
    float* out = (float*)d_out;
    float* ws  = (float*)d_ws;

    float* mean_cls  = ws;              // B*N*C = 3840
    float* mean_task = ws + 3840;       // B*C   = 768
    float* wt        = ws + 4608;       // B*C   = 768
    float* wc        = ws + 5376;       // B*N*C = 3840
    float* ns        = ws + 9216;       // B*N*T = 1960
    float* xs_rw     = ws + 11176;      // B*N*C = 3840
    float* xsn       = ws + 15016;      // B*N*C = 3840

    k_mean_cls<<<B_ * N_, C_, 0, stream>>>(feat_shot, mean_cls);
    k_mean_task<<<B_, C_, 0, stream>>>(mean_cls, mean_task);
    k_mlp<<<B_, 512, 0, stream>>>(mean_task, w1_task, b1_task, w2_task, b2_task, wt, 0);
    k_mlp<<<B_ * N_, 512, 0, stream>>>(mean_cls, w1_cls, b1_cls, w2_cls, b2_cls, wc, 1);
    k_shot<<<B_ * N_, 512, 0, stream>>>(feat_shot, x_shot, wt, wc, ns, xs_rw, xsn);

    size_t lds = (size_t)(T_ * C_ + T_ * PAD_ + C_ * PAD_ + 16 * N_ + 2 * N_) * sizeof(float);
    k_main<<<B_ * Q_, 512, lds, stream>>>(feat_shot, feat_query, x_query,
                                          wt, wc, ns, xs_rw, xsn, out);
}